// DVHGlobalLoss_68315749810521
// MI455X (gfx1250) — compile-verified
//
#include <hip/hip_runtime.h>

// ---------------------------------------------------------------------------
// DVH global loss for MI455X (gfx1250, wave32).
// Pass 1: memory-bound masked histogramming (LDS-privatized uint bins).
// Pass 2: suffix-sum via v_wmma_f32_16x16x4_f32 (D = T x G, T strictly upper
//         triangular ones), then MSE reduction to a scalar.
// ---------------------------------------------------------------------------

typedef __attribute__((ext_vector_type(2))) float v2f;
typedef __attribute__((ext_vector_type(8))) float v8f;

#define NBATCH 4
#define NVOX   (128 * 128 * 128)   // 2,097,152 voxels per batch
#define HBINS  512                 // 501 bins padded to 512
#define NREAL  501                 // searchsorted result range [0, 500]

// c = #{ j in [0,500) : bins[j] <= x },  bins[j] = j * 75/499
__device__ __forceinline__ int dose_bin(float x) {
    int c = (int)(x * (499.0f / 75.0f)) + 1;  // trunc == floor for x >= 0
    if (x < 0.0f) c = 0;
    if (c > 500) c = 500;
    return c;
}

__global__ void zeroKernel(unsigned* __restrict__ p, int n) {
    int i = blockIdx.x * blockDim.x + threadIdx.x;
    if (i < n) p[i] = 0u;
}

// grid = (blocksPerBatch, NBATCH), block = 256
__global__ void histKernel(const float* __restrict__ dpred,
                           const float* __restrict__ dgt,
                           const float* __restrict__ mask,
                           unsigned* __restrict__ gh) {
    __shared__ unsigned hP[HBINS];
    __shared__ unsigned hG[HBINS];
    for (int i = threadIdx.x; i < HBINS; i += blockDim.x) { hP[i] = 0u; hG[i] = 0u; }
    __syncthreads();

    const int b = blockIdx.y;
    const float4* __restrict__ p4 = (const float4*)(dpred + (size_t)b * NVOX);
    const float4* __restrict__ g4 = (const float4*)(dgt   + (size_t)b * NVOX);
    const float4* __restrict__ m4 = (const float4*)(mask  + (size_t)b * NVOX);

    const int n4     = NVOX / 4;                 // 524288 float4 per batch
    const int stride = gridDim.x * blockDim.x;   // grid-stride in float4 units

    for (int i = blockIdx.x * blockDim.x + threadIdx.x; i < n4; i += stride) {
        // prefetch next grid-stride tile (speculative; OOB silently dropped)
        __builtin_prefetch(p4 + i + stride, 0, 0);
        __builtin_prefetch(g4 + i + stride, 0, 0);
        __builtin_prefetch(m4 + i + stride, 0, 0);

        float4 mv = m4[i];
        float4 pv = p4[i];
        float4 gv = g4[i];
        // mask is exactly 0.0 or 1.0 -> integer counting is exact
        if (mv.x != 0.0f) { atomicAdd(&hP[dose_bin(pv.x)], 1u); atomicAdd(&hG[dose_bin(gv.x)], 1u); }
        if (mv.y != 0.0f) { atomicAdd(&hP[dose_bin(pv.y)], 1u); atomicAdd(&hG[dose_bin(gv.y)], 1u); }
        if (mv.z != 0.0f) { atomicAdd(&hP[dose_bin(pv.z)], 1u); atomicAdd(&hG[dose_bin(gv.z)], 1u); }
        if (mv.w != 0.0f) { atomicAdd(&hP[dose_bin(pv.w)], 1u); atomicAdd(&hG[dose_bin(gv.w)], 1u); }
    }
    __syncthreads();

    unsigned* ghP = gh + (size_t)b * HBINS;
    unsigned* ghG = gh + (size_t)(NBATCH + b) * HBINS;
    for (int i = threadIdx.x; i < HBINS; i += blockDim.x) {
        unsigned cp = hP[i], cg = hG[i];
        if (cp) atomicAdd(&ghP[i], cp);
        if (cg) atomicAdd(&ghG[i], cg);
    }
}

// One workgroup, 128 threads = 4 wave32's; wave w handles batch w.
// Computes strict suffix sums of g = (histP - histG)/denom via WMMA, then MSE.
__global__ void dvhReduceKernel(const unsigned* __restrict__ gh,
                                float* __restrict__ out) {
    __shared__ float sg  [NBATCH][HBINS];  // scaled diff histogram (padded)
    __shared__ float ssuf[NBATCH][HBINS];  // within-chunk strict suffix sums
    __shared__ float stot[NBATCH][32];     // per-chunk totals
    __shared__ float scs [NBATCH][32];     // strict suffix over chunk totals
    __shared__ float spart[NBATCH];

    const int lane = threadIdx.x & 31;
    const int b    = threadIdx.x >> 5;

    // ---- load histograms, compute denom = sum(mask) = sum(histP) ----
    unsigned localSum = 0u;
    for (int i = lane; i < HBINS; i += 32) {
        unsigned hp = (i < NREAL) ? gh[(size_t)b * HBINS + i] : 0u;
        unsigned hg = (i < NREAL) ? gh[(size_t)(NBATCH + b) * HBINS + i] : 0u;
        localSum += hp;
        sg[b][i] = (float)(int)hp - (float)(int)hg;
    }
    for (int off = 16; off > 0; off >>= 1) localSum += __shfl_down(localSum, off, 32);
    unsigned denomU = (unsigned)__shfl((int)localSum, 0, 32);
    const float scale = 1.0f / ((float)denomU + 1e-6f);
    for (int i = lane; i < HBINS; i += 32) sg[b][i] *= scale;
    __syncthreads();

    // ---- within-chunk strict suffix sums: D = T(16x16, T[r][c]=c>r) x G ----
    // G[r][q] = sg[16*q + r]; two 16-column halves of the 16x32 matrix.
    const int mrow = lane & 15;                // A-matrix row M (both lane halves)
    const int kofs = (lane < 16) ? 0 : 2;      // K offset per lane half
    const int rofs = (lane < 16) ? 0 : 8;      // D-row offset per lane half
    for (int h = 0; h < 2; ++h) {
        const int col = 16 * h + mrow;         // B/D column N handled by this lane
        v8f acc = {};                          // C = 0
#pragma unroll
        for (int kk = 0; kk < 16; kk += 4) {
            const int k0 = kk + kofs;
            v2f a, bb;
            a.x  = (k0     > mrow) ? 1.0f : 0.0f;   // T[mrow][k0]
            a.y  = (k0 + 1 > mrow) ? 1.0f : 0.0f;   // T[mrow][k0+1]
            bb.x = sg[b][16 * col + k0];            // G[k0][col]
            bb.y = sg[b][16 * col + k0 + 1];        // G[k0+1][col]
            acc = __builtin_amdgcn_wmma_f32_16x16x4_f32(
                false, a, false, bb, (short)0, acc, false, false);
        }
#pragma unroll
        for (int v = 0; v < 8; ++v)
            ssuf[b][16 * col + rofs + v] = acc[v];  // D[rofs+v][col]
    }
    __syncthreads();

    // ---- chunk totals: total[q] = strict-suffix-at-0 + g[16q] ----
    stot[b][lane] = ssuf[b][16 * lane] + sg[b][16 * lane];
    __syncthreads();

    // ---- strict suffix over the 32 chunk totals (tiny) ----
    float cs = 0.0f;
    for (int j = lane + 1; j < 32; ++j) cs += stot[b][j];
    scs[b][lane] = cs;
    __syncthreads();

    // ---- MSE: diff[k] = sum_{c >= k+1} g[c] = ssuf[k] + scs[k>>4] ----
    float acc2 = 0.0f;
    for (int k = lane; k < 500; k += 32) {
        float d = ssuf[b][k] + scs[b][k >> 4];
        acc2 += d * d;
    }
    for (int off = 16; off > 0; off >>= 1) acc2 += __shfl_down(acc2, off, 32);
    if (lane == 0) spart[b] = acc2;
    __syncthreads();
    if (threadIdx.x == 0)
        out[0] = (spart[0] + spart[1] + spart[2] + spart[3]) * (1.0f / 2000.0f);
}

extern "C" void kernel_launch(void* const* d_in, const int* in_sizes, int n_in,
                              void* d_out, int out_size, void* d_ws, size_t ws_size,
                              hipStream_t stream) {
    const float* d_pred = (const float*)d_in[0];
    const float* d_gt   = (const float*)d_in[1];
    const float* d_mask = (const float*)d_in[2];
    float*       out    = (float*)d_out;
    unsigned*    gh     = (unsigned*)d_ws;     // [2][NBATCH][HBINS] uint counters

    const int nCounters = 2 * NBATCH * HBINS;  // 4096
    zeroKernel<<<(nCounters + 255) / 256, 256, 0, stream>>>(gh, nCounters);

    dim3 gridA(256, NBATCH, 1);                // 1024 WGs, 8 float4 per thread
    histKernel<<<gridA, 256, 0, stream>>>(d_pred, d_gt, d_mask, gh);

    dvhReduceKernel<<<1, 128, 0, stream>>>(gh, out);
}